// DGCNN_encoder_38319698215627
// MI455X (gfx1250) — compile-verified
//
#include <hip/hip_runtime.h>
#include <math.h>

#define EPSV 1e-5f
#define NEG_INF (-3.402823466e38f)
#define POS_INF (3.402823466e38f)

typedef __attribute__((ext_vector_type(2))) float v2f;
typedef __attribute__((ext_vector_type(8))) float v8f;

// ======================= tnet(3) kernels =======================

// out[b][o][n] = bias[o] + sum_c W[o][c] * in[b][c][n]
__global__ void k_conv1d(float* __restrict__ out, const float* __restrict__ in,
                         const float* __restrict__ W, const float* __restrict__ bias,
                         int Cin, int Cout) {
  int b = blockIdx.x / Cout;
  int o = blockIdx.x % Cout;
  const float* wrow = W + (size_t)o * Cin;
  const float* inb = in + (size_t)b * Cin * 1024;
  float* outb = out + ((size_t)b * Cout + o) * 1024;
  float bia = bias[o];
  for (int n = threadIdx.x; n < 1024; n += blockDim.x) {
    float acc = bia;
    for (int c = 0; c < Cin; ++c) acc += wrow[c] * inb[c * 1024 + n];
    outb[n] = acc;
  }
}

// per-channel mean/var over (b,n) -> scale/shift  (training-mode BN)
__global__ void k_chan_stats(const float* __restrict__ buf, int C,
                             const float* __restrict__ g, const float* __restrict__ be,
                             float* __restrict__ ss) {
  __shared__ float rs[256], rs2[256];
  int o = blockIdx.x;
  float s = 0.f, s2 = 0.f;
  for (int t = threadIdx.x; t < 8192; t += 256) {
    int b = t >> 10, n = t & 1023;
    float v = buf[((size_t)b * C + o) * 1024 + n];
    s += v; s2 += v * v;
  }
  rs[threadIdx.x] = s; rs2[threadIdx.x] = s2;
  __syncthreads();
  for (int st = 128; st > 0; st >>= 1) {
    if (threadIdx.x < st) { rs[threadIdx.x] += rs[threadIdx.x + st]; rs2[threadIdx.x] += rs2[threadIdx.x + st]; }
    __syncthreads();
  }
  if (threadIdx.x == 0) {
    float mean = rs[0] / 8192.f;
    float var  = rs2[0] / 8192.f - mean * mean;
    float sc = g[o] * rsqrtf(var + EPSV);
    ss[2 * o] = sc;
    ss[2 * o + 1] = be[o] - mean * sc;
  }
}

__global__ void k_bn_relu(float* __restrict__ buf, int C, const float* __restrict__ ss) {
  int b = blockIdx.x / C, o = blockIdx.x % C;
  float sc = ss[2 * o], sh = ss[2 * o + 1];
  float* p = buf + ((size_t)b * C + o) * 1024;
  for (int n = threadIdx.x; n < 1024; n += blockDim.x) {
    float v = sc * p[n] + sh;
    p[n] = v > 0.f ? v : 0.f;
  }
}

// conv2 (128 -> 1024 ch) fused with per-(b,o) sum/sumsq/max/min over n
__global__ void k_conv2_reduce(const float* __restrict__ h1, const float* __restrict__ W2,
                               const float* __restrict__ b2,
                               float* __restrict__ sum_bo, float* __restrict__ sumsq_bo,
                               float* __restrict__ max_bo, float* __restrict__ min_bo) {
  __shared__ float sW[128];
  __shared__ float rs[128], rs2[128], rmx[128], rmn[128];
  int b = blockIdx.x >> 10, o = blockIdx.x & 1023;
  if (threadIdx.x < 128) sW[threadIdx.x] = W2[(size_t)o * 128 + threadIdx.x];
  __syncthreads();
  const float* hb = h1 + (size_t)b * 128 * 1024;
  float bia = b2[o];
  float s = 0.f, s2 = 0.f, mx = NEG_INF, mn = POS_INF;
  for (int n = threadIdx.x; n < 1024; n += 128) {
    float acc = bia;
    for (int c = 0; c < 128; ++c) acc += sW[c] * hb[c * 1024 + n];
    s += acc; s2 += acc * acc;
    mx = fmaxf(mx, acc); mn = fminf(mn, acc);
  }
  rs[threadIdx.x] = s; rs2[threadIdx.x] = s2; rmx[threadIdx.x] = mx; rmn[threadIdx.x] = mn;
  __syncthreads();
  for (int st = 64; st > 0; st >>= 1) {
    if (threadIdx.x < st) {
      rs[threadIdx.x]  += rs[threadIdx.x + st];
      rs2[threadIdx.x] += rs2[threadIdx.x + st];
      rmx[threadIdx.x]  = fmaxf(rmx[threadIdx.x], rmx[threadIdx.x + st]);
      rmn[threadIdx.x]  = fminf(rmn[threadIdx.x], rmn[threadIdx.x + st]);
    }
    __syncthreads();
  }
  if (threadIdx.x == 0) {
    int i = b * 1024 + o;
    sum_bo[i] = rs[0]; sumsq_bo[i] = rs2[0]; max_bo[i] = rmx[0]; min_bo[i] = rmn[0];
  }
}

// finalize BN(affine)+relu+max-over-n using monotonicity: pooled[b][o]
__global__ void k_pool_bn(const float* __restrict__ sum_bo, const float* __restrict__ sumsq_bo,
                          const float* __restrict__ max_bo, const float* __restrict__ min_bo,
                          const float* __restrict__ g, const float* __restrict__ be,
                          float* __restrict__ pooled) {
  int o = blockIdx.x * blockDim.x + threadIdx.x;
  if (o >= 1024) return;
  float s = 0.f, s2 = 0.f;
  for (int b = 0; b < 8; ++b) { s += sum_bo[b * 1024 + o]; s2 += sumsq_bo[b * 1024 + o]; }
  float mean = s / 8192.f, var = s2 / 8192.f - mean * mean;
  float sc = g[o] * rsqrtf(var + EPSV), sh = be[o] - mean * sc;
  for (int b = 0; b < 8; ++b) {
    float m = (sc >= 0.f) ? max_bo[b * 1024 + o] : min_bo[b * 1024 + o];
    float y = sc * m + sh;
    pooled[b * 1024 + o] = y > 0.f ? y : 0.f;
  }
}

// fc + BN(batch of 8) + relu
__global__ void k_fc_bn_relu(const float* __restrict__ in, const float* __restrict__ W,
                             const float* __restrict__ bias, const float* __restrict__ g,
                             const float* __restrict__ be, float* __restrict__ out,
                             int Cin, int Cout) {
  __shared__ float red[8][128];
  int f = blockIdx.x;
  float acc[8];
#pragma unroll
  for (int b = 0; b < 8; ++b) acc[b] = 0.f;
  for (int c = threadIdx.x; c < Cin; c += 128) {
    float w = W[(size_t)f * Cin + c];
#pragma unroll
    for (int b = 0; b < 8; ++b) acc[b] += w * in[b * Cin + c];
  }
#pragma unroll
  for (int b = 0; b < 8; ++b) red[b][threadIdx.x] = acc[b];
  __syncthreads();
  for (int st = 64; st > 0; st >>= 1) {
    if (threadIdx.x < st)
#pragma unroll
      for (int b = 0; b < 8; ++b) red[b][threadIdx.x] += red[b][threadIdx.x + st];
    __syncthreads();
  }
  if (threadIdx.x == 0) {
    float y[8], m = 0.f;
    for (int b = 0; b < 8; ++b) { y[b] = red[b][0] + bias[f]; m += y[b]; }
    m *= 0.125f;
    float v = 0.f;
    for (int b = 0; b < 8; ++b) { float d = y[b] - m; v += d * d; }
    v *= 0.125f;
    float sc = g[f] * rsqrtf(v + EPSV);
    for (int b = 0; b < 8; ++b) {
      float r = sc * (y[b] - m) + be[f];
      out[b * Cout + f] = r > 0.f ? r : 0.f;
    }
  }
}

// fc2 + bias + eye(3)   x1[b][j], j in 0..8
__global__ void k_fc_final(const float* __restrict__ h4, const float* __restrict__ Wf2,
                           const float* __restrict__ bf2, float* __restrict__ x1) {
  __shared__ float r[64];
  int b = blockIdx.x / 9, j = blockIdx.x % 9;
  float s = 0.f;
  for (int c = threadIdx.x; c < 256; c += 64) s += Wf2[j * 256 + c] * h4[b * 256 + c];
  r[threadIdx.x] = s;
  __syncthreads();
  for (int st = 32; st > 0; st >>= 1) {
    if (threadIdx.x < st) r[threadIdx.x] += r[threadIdx.x + st];
    __syncthreads();
  }
  if (threadIdx.x == 0) x1[b * 9 + j] = r[0] + bf2[j] + ((j % 4) == 0 ? 1.f : 0.f);
}

// new_out[b][i][n] = sum_j x1[b][i][j] * x[b][j][n]
__global__ void k_apply_t3(const float* __restrict__ x, const float* __restrict__ x1,
                           float* __restrict__ new_out) {
  int b = blockIdx.x / 3, i = blockIdx.x % 3;
  float a0 = x1[b * 9 + i * 3 + 0], a1 = x1[b * 9 + i * 3 + 1], a2 = x1[b * 9 + i * 3 + 2];
  const float* xb = x + (size_t)b * 3 * 1024;
  float* ob = new_out + ((size_t)b * 3 + i) * 1024;
  for (int n = threadIdx.x; n < 1024; n += blockDim.x)
    ob[n] = a0 * xb[n] + a1 * xb[1024 + n] + a2 * xb[2048 + n];
}

// ======================= EdgeConv kernels =======================

__global__ void k_sqnorm(const float* __restrict__ xin, int C, int bStride,
                         float* __restrict__ sq) {
  int t = blockIdx.x * 256 + threadIdx.x;   // 0..8191
  int b = t >> 10, n = t & 1023;
  const float* p = xin + (size_t)b * bStride + n;
  float s = 0.f;
  for (int c = 0; c < C; ++c) { float v = p[c * 1024]; s += v * v; }
  sq[t] = s;
}

// top-20 argmax rounds over pd (lower index wins ties, matches lax.top_k)
__global__ void k_knn(const float* __restrict__ xin, int C, int bStride,
                      const float* __restrict__ sq, int* __restrict__ idxOut) {
  __shared__ float pd[1024];
  __shared__ float xc[128];
  __shared__ float bestV[256];
  __shared__ int   bestI[256];
  int bn = blockIdx.x;
  int b = bn >> 10, n = bn & 1023;
  const float* xb = xin + (size_t)b * bStride;
  for (int c = threadIdx.x; c < C; c += 256) xc[c] = xb[c * 1024 + n];
  __syncthreads();
  float sqn = sq[b * 1024 + n];
  for (int m = threadIdx.x; m < 1024; m += 256) {
    float d = 0.f;
    for (int c = 0; c < C; ++c) d += xc[c] * xb[c * 1024 + m];
    pd[m] = 2.f * d - sqn - sq[b * 1024 + m];
  }
  __syncthreads();
  for (int r = 0; r < 20; ++r) {
    float bv = NEG_INF; int bi = 1 << 30;
    for (int m = threadIdx.x; m < 1024; m += 256) {
      float v = pd[m];
      if (v > bv || (v == bv && m < bi)) { bv = v; bi = m; }
    }
    bestV[threadIdx.x] = bv; bestI[threadIdx.x] = bi;
    __syncthreads();
    for (int st = 128; st > 0; st >>= 1) {
      if (threadIdx.x < st) {
        float ov = bestV[threadIdx.x + st]; int oi = bestI[threadIdx.x + st];
        if (ov > bestV[threadIdx.x] || (ov == bestV[threadIdx.x] && oi < bestI[threadIdx.x])) {
          bestV[threadIdx.x] = ov; bestI[threadIdx.x] = oi;
        }
      }
      __syncthreads();
    }
    if (threadIdx.x == 0) {
      idxOut[(size_t)bn * 20 + r] = bestI[0];
      pd[bestI[0]] = NEG_INF;
    }
    __syncthreads();
  }
}

// Per point: Y(Oxk) = Wn(OxC) * Xnbr(Cxk) + [(Wc-Wn)*xc + b]  via V_WMMA_F32_16X16X4_F32.
// Both operands staged in LDS (zero-padded), so the K-loop is pure ds_load + v_wmma,
// no branches / global waits. Then leaky_relu(0.1), per-(b,o,n) max/min over k and
// per-point partial sum/sumsq for the training-mode BN statistics.
__global__ void k_edge_wmma(const float* __restrict__ xin, int C, int bStride,
                            const float* __restrict__ W, const float* __restrict__ bias,
                            const int* __restrict__ idx,
                            float* __restrict__ ymax, float* __restrict__ ymin,
                            float* __restrict__ psum, float* __restrict__ psumsq, int O) {
  extern __shared__ float smem[];
  const int Cpad = (C + 3) & ~3;
  const int WnStride = Cpad + 1;          // odd stride: consecutive rows hit distinct banks
  float* sWn  = smem;                     // O x WnStride (Wn, zero-padded to Cpad)
  float* Xn   = sWn + (size_t)O * WnStride; // Cpad x 33 (neighbor cols 0..31, padded stride)
  float* cctr = Xn + (size_t)Cpad * 33;   // O
  float* xcs  = cctr + O;                 // C
  int*   sIdx = (int*)(xcs + C);          // 32
  const int tid = threadIdx.x, bd = blockDim.x;
  const int bn = blockIdx.x;
  const int b = bn >> 10, n = bn & 1023;
  const float* xb = xin + (size_t)b * bStride;

  if (tid < 32) sIdx[tid] = (tid < 20) ? idx[(size_t)bn * 20 + tid] : 0;
  for (int c = tid; c < C; c += bd) xcs[c] = xb[c * 1024 + n];
  // stage Wn tile (first C columns of each 2C-wide row of W), zero-padded to Cpad
  for (int t = tid; t < O * Cpad; t += bd) {
    int o = t / Cpad, c = t - o * Cpad;
    sWn[o * WnStride + c] = (c < C) ? W[(size_t)o * 2 * C + c] : 0.f;
  }
  __syncthreads();

  // gather neighbor columns (zero-pad cols 20..31 and rows C..Cpad-1)
  for (int t = tid; t < Cpad * 32; t += bd) {
    int ci = t >> 5, kk = t & 31;
    float v = 0.f;
    if (ci < C && kk < 20) v = xb[ci * 1024 + sIdx[kk]];
    Xn[ci * 33 + kk] = v;
  }
  // center term: (Wc - Wn) . xc + bias
  for (int o = tid; o < O; o += bd) {
    const float* wr = W + (size_t)o * 2 * C;
    float acc = bias[o];
    for (int c = 0; c < C; ++c) acc += (wr[C + c] - wr[c]) * xcs[c];
    cctr[o] = acc;
  }
  __syncthreads();

  const int lane = tid & 31;
  const int hi = lane >> 4;         // lane half
  const int ml = lane & 15;
  const int obase = (tid >> 5) * 16; // one 16-row O-tile per wave

  v8f acc0 = {0.f, 0.f, 0.f, 0.f, 0.f, 0.f, 0.f, 0.f};
  v8f acc1 = {0.f, 0.f, 0.f, 0.f, 0.f, 0.f, 0.f, 0.f};
  const float* wa = sWn + (size_t)(obase + ml) * WnStride;  // A row M=ml

#pragma unroll 2
  for (int cb = 0; cb < Cpad; cb += 4) {
    int c0 = cb + 2 * hi;
    v2f a;                                     // A: VGPR0 = K{0,2}, VGPR1 = K{1,3}
    a.x = wa[c0];
    a.y = wa[c0 + 1];
    v2f b0v, b1v;                              // B: rows K{0,2}/{1,3}, cols = lane%16
    b0v.x = Xn[c0 * 33 + ml];        b0v.y = Xn[(c0 + 1) * 33 + ml];
    b1v.x = Xn[c0 * 33 + 16 + ml];   b1v.y = Xn[(c0 + 1) * 33 + 16 + ml];
    acc0 = __builtin_amdgcn_wmma_f32_16x16x4_f32(false, a, false, b0v, (short)0, acc0, false, false);
    acc1 = __builtin_amdgcn_wmma_f32_16x16x4_f32(false, a, false, b1v, (short)0, acc1, false, false);
  }

  // epilogue: leaky relu, mask k>=20, butterfly-reduce the 16 k-lanes per half
  for (int v = 0; v < 8; ++v) {
    int o = obase + v + 8 * hi;          // D row M = v + 8*hi
    float cadd = cctr[o];
    float y0 = acc0[v] + cadd; y0 = (y0 >= 0.f) ? y0 : 0.1f * y0;   // k = ml (<20 always)
    float y1 = acc1[v] + cadd; y1 = (y1 >= 0.f) ? y1 : 0.1f * y1;   // k = 16+ml
    bool ok1 = (ml < 4);
    float s  = y0 + (ok1 ? y1 : 0.f);
    float s2 = y0 * y0 + (ok1 ? y1 * y1 : 0.f);
    float mx = ok1 ? fmaxf(y0, y1) : y0;
    float mn = ok1 ? fminf(y0, y1) : y0;
    for (int d = 1; d < 16; d <<= 1) {
      s  += __shfl_xor(s, d);
      s2 += __shfl_xor(s2, d);
      mx = fmaxf(mx, __shfl_xor(mx, d));
      mn = fminf(mn, __shfl_xor(mn, d));
    }
    if (ml == 0) {
      ymax[((size_t)b * O + o) * 1024 + n] = mx;
      ymin[((size_t)b * O + o) * 1024 + n] = mn;
      psum[(size_t)bn * O + o] = s;
      psumsq[(size_t)bn * O + o] = s2;
    }
  }
}

// reduce per-point partials -> channel scale/shift (count = 8*1024*20)
__global__ void k_edge_stats(const float* __restrict__ psum, const float* __restrict__ psumsq,
                             const float* __restrict__ g, const float* __restrict__ be,
                             float* __restrict__ ss, int O) {
  __shared__ float rs[256], rs2[256];
  int o = blockIdx.x;
  float s = 0.f, s2 = 0.f;
  for (int t = threadIdx.x; t < 8192; t += 256) {
    s  += psum[(size_t)t * O + o];
    s2 += psumsq[(size_t)t * O + o];
  }
  rs[threadIdx.x] = s; rs2[threadIdx.x] = s2;
  __syncthreads();
  for (int st = 128; st > 0; st >>= 1) {
    if (threadIdx.x < st) { rs[threadIdx.x] += rs[threadIdx.x + st]; rs2[threadIdx.x] += rs2[threadIdx.x + st]; }
    __syncthreads();
  }
  if (threadIdx.x == 0) {
    const float inv = 1.f / 163840.f;
    float mean = rs[0] * inv;
    float var  = rs2[0] * inv - mean * mean;
    float sc = g[o] * rsqrtf(var + EPSV);
    ss[2 * o] = sc;
    ss[2 * o + 1] = be[o] - mean * sc;
  }
}

// max_k(BN(y)) via affine monotonicity; write into concat slice of d_out
__global__ void k_edge_final(const float* __restrict__ ymax, const float* __restrict__ ymin,
                             const float* __restrict__ ss, float* __restrict__ out,
                             int O, int outBase) {
  int b = blockIdx.x / O, o = blockIdx.x % O;
  float sc = ss[2 * o], sh = ss[2 * o + 1];
  const float* mp = (sc >= 0.f) ? ymax : ymin;
  const float* src = mp + ((size_t)b * O + o) * 1024;
  float* dst = out + ((size_t)b * 512 + outBase + o) * 1024;
  for (int n = threadIdx.x; n < 1024; n += blockDim.x)
    dst[n] = sc * src[n] + sh;
}

// ======================= host driver =======================

extern "C" void kernel_launch(void* const* d_in, const int* in_sizes, int n_in,
                              void* d_out, int out_size, void* d_ws, size_t ws_size,
                              hipStream_t stream) {
  (void)in_sizes; (void)n_in; (void)out_size; (void)ws_size;
  const float* x = (const float*)d_in[0];
  // t3 params (insertion-order DFS flatten of setup_inputs dict)
  const float* t3W0  = (const float*)d_in[1];
  const float* t3b0  = (const float*)d_in[2];
  const float* t3g0  = (const float*)d_in[3];
  const float* t3be0 = (const float*)d_in[4];
  const float* t3W1  = (const float*)d_in[5];
  const float* t3b1  = (const float*)d_in[6];
  const float* t3g1  = (const float*)d_in[7];
  const float* t3be1 = (const float*)d_in[8];
  const float* t3W2  = (const float*)d_in[9];
  const float* t3b2  = (const float*)d_in[10];
  const float* t3g2  = (const float*)d_in[11];
  const float* t3be2 = (const float*)d_in[12];
  const float* Wf0   = (const float*)d_in[13];
  const float* bf0   = (const float*)d_in[14];
  const float* g3    = (const float*)d_in[15];
  const float* be3   = (const float*)d_in[16];
  const float* Wf1   = (const float*)d_in[17];
  const float* bf1   = (const float*)d_in[18];
  const float* g4    = (const float*)d_in[19];
  const float* be4   = (const float*)d_in[20];
  const float* Wf2   = (const float*)d_in[21];
  const float* bf2   = (const float*)d_in[22];
  // indices 23..44 = t64 (unused), 45..56 = tr conv params (unused)
  const float* ecW[4]  = {(const float*)d_in[57], (const float*)d_in[61], (const float*)d_in[65], (const float*)d_in[69]};
  const float* ecB[4]  = {(const float*)d_in[58], (const float*)d_in[62], (const float*)d_in[66], (const float*)d_in[70]};
  const float* ecG[4]  = {(const float*)d_in[59], (const float*)d_in[63], (const float*)d_in[67], (const float*)d_in[71]};
  const float* ecBE[4] = {(const float*)d_in[60], (const float*)d_in[64], (const float*)d_in[68], (const float*)d_in[72]};

  float* ws = (float*)d_ws;
  size_t off = 0;
  auto alloc = [&](size_t nf) { float* p = ws + off; off += nf; return p; };
  float* h0       = alloc((size_t)8 * 64 * 1024);
  float* h1       = alloc((size_t)8 * 128 * 1024);
  float* ss       = alloc(2 * 1024);
  float* sum_bo   = alloc(8192);
  float* sumsq_bo = alloc(8192);
  float* max_bo   = alloc(8192);
  float* min_bo   = alloc(8192);
  float* pooled   = alloc(8192);
  float* h3       = alloc(8 * 512);
  float* h4       = alloc(8 * 256);
  float* x1buf    = alloc(72);
  float* new_out  = alloc((size_t)8 * 3 * 1024);
  float* sq       = alloc(8192);
  int*   idx      = (int*)alloc((size_t)8192 * 20);
  float* ymax     = alloc((size_t)8 * 256 * 1024);
  float* ymin     = alloc((size_t)8 * 256 * 1024);
  float* psum     = alloc((size_t)8192 * 256);
  float* psumsq   = alloc((size_t)8192 * 256);

  // ---- tnet(x, t3, 3) -> x1 ----
  k_conv1d<<<8 * 64, 256, 0, stream>>>(h0, x, t3W0, t3b0, 3, 64);
  k_chan_stats<<<64, 256, 0, stream>>>(h0, 64, t3g0, t3be0, ss);
  k_bn_relu<<<8 * 64, 256, 0, stream>>>(h0, 64, ss);
  k_conv1d<<<8 * 128, 256, 0, stream>>>(h1, h0, t3W1, t3b1, 64, 128);
  k_chan_stats<<<128, 256, 0, stream>>>(h1, 128, t3g1, t3be1, ss);
  k_bn_relu<<<8 * 128, 256, 0, stream>>>(h1, 128, ss);
  k_conv2_reduce<<<8192, 128, 0, stream>>>(h1, t3W2, t3b2, sum_bo, sumsq_bo, max_bo, min_bo);
  k_pool_bn<<<4, 256, 0, stream>>>(sum_bo, sumsq_bo, max_bo, min_bo, t3g2, t3be2, pooled);
  k_fc_bn_relu<<<512, 128, 0, stream>>>(pooled, Wf0, bf0, g3, be3, h3, 1024, 512);
  k_fc_bn_relu<<<256, 128, 0, stream>>>(h3, Wf1, bf1, g4, be4, h4, 512, 256);
  k_fc_final<<<72, 64, 0, stream>>>(h4, Wf2, bf2, x1buf);
  k_apply_t3<<<24, 256, 0, stream>>>(x, x1buf, new_out);

  float* dout = (float*)d_out;
  auto run_edge = [&](const float* xin, int C, int bStride, int li, int O, int outBase) {
    int Cpad = (C + 3) & ~3;
    size_t smem = (size_t)((size_t)O * (Cpad + 1) + (size_t)Cpad * 33 + O + C + 32) * sizeof(float);
    k_sqnorm<<<32, 256, 0, stream>>>(xin, C, bStride, sq);
    k_knn<<<8192, 256, 0, stream>>>(xin, C, bStride, sq, idx);
    k_edge_wmma<<<8192, 2 * O, smem, stream>>>(xin, C, bStride, ecW[li], ecB[li], idx,
                                               ymax, ymin, psum, psumsq, O);
    k_edge_stats<<<O, 256, 0, stream>>>(psum, psumsq, ecG[li], ecBE[li], ss, O);
    k_edge_final<<<8 * O, 256, 0, stream>>>(ymax, ymin, ss, dout, O, outBase);
  };

  // o1..o4, each written directly into its concat slice; next layer reads from d_out
  run_edge(new_out,             3,   3 * 1024,   0,  64,   0);
  run_edge(dout + 0 * 1024,     64,  512 * 1024, 1,  64,  64);
  run_edge(dout + 64 * 1024,    64,  512 * 1024, 2, 128, 128);
  run_edge(dout + 128 * 1024,   128, 512 * 1024, 3, 256, 256);
}